// SpecialSpmm_83734682402939
// MI455X (gfx1250) — compile-verified
//
#include <hip/hip_runtime.h>
#include <stdint.h>

// COO SpMM scatter: out[row[e],:] += values[e] * b[col[e],:]
// N=100000, E=1600000, D=128.  Memory/atomic bound; b (51MB) and out (51MB)
// both fit in the 192MB L2, so the gather/scatter streams are L2-resident.
// CDNA5 paths: async global->LDS staging of edge metadata (ASYNCcnt, double
// buffered, s_wait_asynccnt) + no-return global_atomic_add_f32 scatter.
// Edge metadata is wave-uniform -> readfirstlane to SGPRs so gather/scatter
// use saddr addressing (address math on SALU, co-issued with VALU).

#define DCOLS 128
#define WPB   8     // waves per block (wave32)
#define TPB   256

__device__ __forceinline__ void async_b64(void* lds, const void* g) {
  asm volatile("global_load_async_to_lds_b64 %0, %1, off"
               :: "v"((__attribute__((address_space(3))) void*)lds),
                  "v"(g)
               : "memory");
}
__device__ __forceinline__ void async_b32(void* lds, const void* g) {
  asm volatile("global_load_async_to_lds_b32 %0, %1, off"
               :: "v"((__attribute__((address_space(3))) void*)lds),
                  "v"(g)
               : "memory");
}
__device__ __forceinline__ void wait_async3() {   // oldest 3 async ops done
  asm volatile("s_wait_asynccnt 3" ::: "memory");
}
__device__ __forceinline__ void wait_async0() {   // all async ops done
  asm volatile("s_wait_asynccnt 0" ::: "memory");
}

__global__ __launch_bounds__(TPB) void spmm_coo_scatter(
    const long long* __restrict__ rowIdx,
    const long long* __restrict__ colIdx,
    const float*     __restrict__ vals,
    const float*     __restrict__ bmat,
    float*           __restrict__ out,
    int E)
{
  // Per-wave double-buffered staging area: 32 edges of metadata per batch.
  __shared__ long long s_row[2][WPB][32];   // 4 KB
  __shared__ long long s_col[2][WPB][32];   // 4 KB
  __shared__ float     s_val[2][WPB][32];   // 2 KB

  const int lane   = threadIdx.x & 31;
  const int wv     = threadIdx.x >> 5;
  const int nb     = (E + 31) >> 5;               // 32-edge batches
  const int stride = gridDim.x * WPB;             // total waves
  int bi = blockIdx.x * WPB + wv;                 // wave-uniform batch id

  if (bi >= nb) return;                           // wave-uniform exit

  // Prologue: stage batch bi into buffer 0 (3 async ops: rows, cols, vals).
  {
    long long e  = (long long)bi * 32 + lane;
    long long ec = (e < (long long)E) ? e : (long long)(E - 1);  // clamp tail
    async_b64(&s_row[0][wv][lane], rowIdx + ec);
    async_b64(&s_col[0][wv][lane], colIdx + ec);
    async_b32(&s_val[0][wv][lane], vals   + ec);
  }

  const int col4 = lane << 2;                     // this lane's 4-float slice
  int buf = 0;
  while (bi < nb) {
    const int nxt = bi + stride;
    if (nxt < nb) {
      // Stage the next batch into the other buffer, then wait for the
      // *current* batch only (async loads complete in order; <=3
      // outstanding => the oldest 3 have landed in LDS).
      long long e  = (long long)nxt * 32 + lane;
      long long ec = (e < (long long)E) ? e : (long long)(E - 1);
      async_b64(&s_row[buf ^ 1][wv][lane], rowIdx + ec);
      async_b64(&s_col[buf ^ 1][wv][lane], colIdx + ec);
      async_b32(&s_val[buf ^ 1][wv][lane], vals   + ec);
      wait_async3();
    } else {
      wait_async0();
    }

    const int cnt = min(32, E - (bi << 5));
    const unsigned* rlo = (const unsigned*)&s_row[buf][wv][0]; // int64 low dword
    const unsigned* clo = (const unsigned*)&s_col[buf][wv][0];
    const float*    vv  = &s_val[buf][wv][0];

    // One wave per edge: lane handles 4 consecutive floats of the 128-wide
    // row.  row/col/val are identical across lanes -> force into SGPRs so
    // the gather/scatter use saddr-form addressing (SALU address math).
    #pragma unroll 4
    for (int i = 0; i < cnt; ++i) {
      const unsigned r = (unsigned)__builtin_amdgcn_readfirstlane((int)rlo[2 * i]);
      const unsigned c = (unsigned)__builtin_amdgcn_readfirstlane((int)clo[2 * i]);
      const float    v = __int_as_float(
          __builtin_amdgcn_readfirstlane(__float_as_int(vv[i])));

      const float4 bv = *(const float4*)(bmat + (size_t)c * DCOLS + col4);
      float* o = out + (size_t)r * DCOLS + col4;
      // Fire-and-forget f32 atomics -> global_atomic_add_f32 (no return),
      // resolved in L2; offsets 0/4/8/12 fold into the instruction.
      __hip_atomic_fetch_add(o + 0, v * bv.x, __ATOMIC_RELAXED, __HIP_MEMORY_SCOPE_AGENT);
      __hip_atomic_fetch_add(o + 1, v * bv.y, __ATOMIC_RELAXED, __HIP_MEMORY_SCOPE_AGENT);
      __hip_atomic_fetch_add(o + 2, v * bv.z, __ATOMIC_RELAXED, __HIP_MEMORY_SCOPE_AGENT);
      __hip_atomic_fetch_add(o + 3, v * bv.w, __ATOMIC_RELAXED, __HIP_MEMORY_SCOPE_AGENT);
    }
    buf ^= 1;
    bi = nxt;
  }
}

extern "C" void kernel_launch(void* const* d_in, const int* in_sizes, int n_in,
                              void* d_out, int out_size, void* d_ws, size_t ws_size,
                              hipStream_t stream) {
  (void)n_in; (void)d_ws; (void)ws_size;
  const long long* idx  = (const long long*)d_in[0];  // int64 [2, E]
  const float*     vals = (const float*)d_in[1];      // f32   [E]
  // d_in[2] is the python scalar n (unused; out_size gives N*D)
  const float*     bmat = (const float*)d_in[3];      // f32   [N, 128]
  float*           out  = (float*)d_out;              // f32   [N, 128]

  const int E = in_sizes[1];

  // Zero the output (graph-capture legal, deterministic each replay).
  hipMemsetAsync(d_out, 0, (size_t)out_size * sizeof(float), stream);

  const int nb = (E + 31) / 32;                 // 50,000 batches for E=1.6M
  int blocks = (nb + WPB - 1) / WPB;
  if (blocks > 1024) blocks = 1024;             // 8192 waves, ~6 batches each
  if (blocks < 1) blocks = 1;

  spmm_coo_scatter<<<blocks, TPB, 0, stream>>>(idx, idx + E, vals, bmat, out, E);
}